// GCN_10737418240588
// MI455X (gfx1250) — compile-verified
//
#include <hip/hip_runtime.h>

typedef __attribute__((ext_vector_type(2))) float v2f;
typedef __attribute__((ext_vector_type(8))) float v8f;

// ---------------- small utility kernels ----------------

__global__ void k_zero_u32(unsigned* p, int n) {
  int i = blockIdx.x * blockDim.x + threadIdx.x;
  if (i < n) p[i] = 0u;
}

__global__ void k_copy_u32(const unsigned* __restrict__ a, unsigned* __restrict__ b, int n) {
  int i = blockIdx.x * blockDim.x + threadIdx.x;
  if (i < n) b[i] = a[i];
}

__global__ void k_deg_count(const int* __restrict__ dst, unsigned* __restrict__ cnt, int E) {
  int e = blockIdx.x * blockDim.x + threadIdx.x;
  if (e < E) atomicAdd(&cnt[dst[e]], 1u);
}

__global__ void k_dis(const unsigned* __restrict__ deg, float* __restrict__ dis, int n) {
  int i = blockIdx.x * blockDim.x + threadIdx.x;
  if (i < n) dis[i] = rsqrtf((float)deg[i] + 1.0f);
}

// ---------------- exclusive prefix scan (single block, chunked LDS scan) ----------------
#define SCAN_T 1024
__global__ void k_scan_exclusive(const unsigned* __restrict__ deg, unsigned* __restrict__ rp,
                                 int n) {
  __shared__ unsigned s[SCAN_T];
  const int tid = threadIdx.x;
  unsigned carry = 0;
  for (int base = 0; base < n; base += SCAN_T) {
    int i = base + tid;
    unsigned v = (i < n) ? deg[i] : 0u;
    s[tid] = v;
    __syncthreads();
    for (int off = 1; off < SCAN_T; off <<= 1) {
      unsigned t = (tid >= off) ? s[tid - off] : 0u;
      __syncthreads();
      s[tid] += t;
      __syncthreads();
    }
    if (i < n) rp[i] = carry + s[tid] - v;   // exclusive prefix
    carry += s[SCAN_T - 1];
    __syncthreads();                          // before s[] is overwritten
  }
  if (tid == 0) rp[n] = carry;
}

// ---------------- CSR fill: csrc[slot] = src, grouped by dst ----------------
__global__ void k_fill_csr(const int* __restrict__ src, const int* __restrict__ dst,
                           unsigned* __restrict__ cursor, unsigned* __restrict__ csrc, int E) {
  int e = blockIdx.x * blockDim.x + threadIdx.x;
  if (e < E) {
    unsigned pos = atomicAdd(&cursor[dst[e]], 1u);
    csrc[pos] = (unsigned)src[e];
  }
}

// ---------------- WMMA fp32 GEMM: C[M,N] = A[M,K] * B[K,N] ----------------
// One wave per 16x16 tile. blockDim.x = 32 * (N/16). gridDim.x = M/16.
__global__ void k_gemm_wmma(const float* __restrict__ A, const float* __restrict__ B,
                            float* __restrict__ C, int K, int N) {
  const int lane = threadIdx.x & 31;
  const int wave = threadIdx.x >> 5;
  const int half = lane >> 4;
  const int l15  = lane & 15;
  const int row0 = blockIdx.x * 16;
  const int col0 = wave * 16;

  const float* Ap = A + (size_t)(row0 + l15) * K + 2 * half;
  const float* Bp = B + (size_t)(2 * half) * N + (col0 + l15);

  v8f acc = {};
  for (int k = 0; k < K; k += 4) {
    v2f a = *(const v2f*)(Ap + k);
    v2f b;
    b.x = Bp[(size_t)k * N];
    b.y = Bp[(size_t)k * N + N];
    acc = __builtin_amdgcn_wmma_f32_16x16x4_f32(false, a, false, b, (short)0, acc, false, false);
  }

  float* Cp = C + (size_t)(row0 + half * 8) * N + col0 + l15;
#pragma unroll
  for (int r = 0; r < 8; ++r) Cp[(size_t)r * N] = acc[r];
}

// ---------------- fused gather + self-loop + bias + relu, F = 128 ----------------
// One wave per dst node; each lane owns 4 consecutive features.
__global__ void k_gather128(const unsigned* __restrict__ rp, const unsigned* __restrict__ csrc,
                            const float* __restrict__ dis, const float* __restrict__ h,
                            const float* __restrict__ bias, float* __restrict__ out) {
  const int lane = threadIdx.x & 31;
  const int node = blockIdx.x * (blockDim.x >> 5) + (threadIdx.x >> 5);
  const float dd = dis[node];
  const unsigned lo = rp[node], hi = rp[node + 1];

  float4 acc = *(const float4*)(h + (size_t)node * 128 + lane * 4);
  const float sw = dd * dd;
  acc.x *= sw; acc.y *= sw; acc.z *= sw; acc.w *= sw;

  for (unsigned j = lo; j < hi; ++j) {
    unsigned s = csrc[j];
    float nrm = dis[s] * dd;
    const float4 hv = *(const float4*)(h + (size_t)s * 128 + lane * 4);
    acc.x = fmaf(hv.x, nrm, acc.x);
    acc.y = fmaf(hv.y, nrm, acc.y);
    acc.z = fmaf(hv.z, nrm, acc.z);
    acc.w = fmaf(hv.w, nrm, acc.w);
  }

  const float4 bb = *(const float4*)(bias + lane * 4);
  float4 r;
  r.x = fmaxf(acc.x + bb.x, 0.0f);
  r.y = fmaxf(acc.y + bb.y, 0.0f);
  r.z = fmaxf(acc.z + bb.z, 0.0f);
  r.w = fmaxf(acc.w + bb.w, 0.0f);
  *(float4*)(out + (size_t)node * 128 + lane * 4) = r;
}

// ---------------- fused gather + self-loop + bias + relu, F = 64 ----------------
__global__ void k_gather64(const unsigned* __restrict__ rp, const unsigned* __restrict__ csrc,
                           const float* __restrict__ dis, const float* __restrict__ h,
                           const float* __restrict__ bias, float* __restrict__ out) {
  const int lane = threadIdx.x & 31;
  const int node = blockIdx.x * (blockDim.x >> 5) + (threadIdx.x >> 5);
  const float dd = dis[node];
  const unsigned lo = rp[node], hi = rp[node + 1];

  float2 acc = *(const float2*)(h + (size_t)node * 64 + lane * 2);
  const float sw = dd * dd;
  acc.x *= sw; acc.y *= sw;

  for (unsigned j = lo; j < hi; ++j) {
    unsigned s = csrc[j];
    float nrm = dis[s] * dd;
    const float2 hv = *(const float2*)(h + (size_t)s * 64 + lane * 2);
    acc.x = fmaf(hv.x, nrm, acc.x);
    acc.y = fmaf(hv.y, nrm, acc.y);
  }

  const float2 bb = *(const float2*)(bias + lane * 2);
  float2 r;
  r.x = fmaxf(acc.x + bb.x, 0.0f);
  r.y = fmaxf(acc.y + bb.y, 0.0f);
  *(float2*)(out + (size_t)node * 64 + lane * 2) = r;
}

// ---------------- fused mean-pool + final linear (batch is sorted) ----------------
// One block per graph: binary-search node range, block-reduce 64 features, 2 outputs.
__global__ void k_pool_lin(const float* __restrict__ a2, const int* __restrict__ batch,
                           const float* __restrict__ Wlin, const float* __restrict__ blin,
                           float* __restrict__ out, int Nn) {
  __shared__ int bounds[2];
  __shared__ float red[256];
  __shared__ float tot[64];
  const int g = blockIdx.x;
  const int t = threadIdx.x;

  if (t < 2) {
    int key = g + t;                    // lower_bound(batch, key)
    int lo = 0, hi = Nn;
    while (lo < hi) {
      int mid = (lo + hi) >> 1;
      if (batch[mid] < key) lo = mid + 1; else hi = mid;
    }
    bounds[t] = lo;
  }
  __syncthreads();
  const int lo = bounds[0], hi = bounds[1];

  const int f = t & 63, sub = t >> 6;   // 4 node-lanes x 64 features
  float acc = 0.0f;
  for (int n = lo + sub; n < hi; n += 4) acc += a2[(size_t)n * 64 + f];
  red[t] = acc;
  __syncthreads();
  if (t < 64) tot[t] = red[t] + red[t + 64] + red[t + 128] + red[t + 192];
  __syncthreads();

  if (t < 2) {
    float inv = 1.0f / fmaxf((float)(hi - lo), 1.0f);
    float s = 0.0f;
    for (int ff = 0; ff < 64; ++ff) s += tot[ff] * Wlin[ff * 2 + t];
    out[g * 2 + t] = s * inv + blin[t];
  }
}

// ---------------- launch ----------------

extern "C" void kernel_launch(void* const* d_in, const int* in_sizes, int n_in,
                              void* d_out, int out_size, void* d_ws, size_t ws_size,
                              hipStream_t stream) {
  (void)n_in; (void)out_size; (void)ws_size;

  const float* x    = (const float*)d_in[0];
  const int*   ei   = (const int*)d_in[1];
  const int*   bat  = (const int*)d_in[2];
  const float* W1   = (const float*)d_in[3];
  const float* b1   = (const float*)d_in[4];
  const float* W2   = (const float*)d_in[5];
  const float* b2   = (const float*)d_in[6];
  const float* Wlin = (const float*)d_in[7];
  const float* blin = (const float*)d_in[8];
  float* out = (float*)d_out;

  const int Nn = in_sizes[2];        // 100000 nodes (multiple of 16)
  const int E  = in_sizes[1] / 2;    // 1600000 edges
  const int* src = ei;
  const int* dst = ei + E;

  // Workspace layout (bytes):
  //  [0)            h_buf   : N*128*4 = 51,200,000   (h1, later h2)
  //  [51,200,000)   agg_buf : N*128*4 = 51,200,000   (a1, later a2)
  //  [102,400,000)  deg     : N*4     =    400,000
  //  [102,800,000)  dis     : N*4     =    400,000
  //  [103,200,000)  row_ptr : (N+1)*4 =    400,256 (reserved)
  //  [103,600,256)  cursor  : N*4     =    400,000
  //  [104,000,256)  csr_src : E*4     =  6,400,000
  char* ws = (char*)d_ws;
  float*    h_buf   = (float*)(ws);
  float*    agg_buf = (float*)(ws + 51200000);
  unsigned* deg     = (unsigned*)(ws + 102400000);
  float*    dis     = (float*)(ws + 102800000);
  unsigned* rp      = (unsigned*)(ws + 103200000);
  unsigned* cursor  = (unsigned*)(ws + 103600256);
  unsigned* csrc    = (unsigned*)(ws + 104000256);

  auto cdiv = [](long long a, long long b) -> int { return (int)((a + b - 1) / b); };

  // degree, normalization, CSR-by-dst
  k_zero_u32<<<cdiv(Nn, 256), 256, 0, stream>>>(deg, Nn);
  k_deg_count<<<cdiv(E, 256), 256, 0, stream>>>(dst, deg, E);
  k_dis<<<cdiv(Nn, 256), 256, 0, stream>>>(deg, dis, Nn);
  k_scan_exclusive<<<1, SCAN_T, 0, stream>>>(deg, rp, Nn);
  k_copy_u32<<<cdiv(Nn, 256), 256, 0, stream>>>(rp, cursor, Nn);
  k_fill_csr<<<cdiv(E, 256), 256, 0, stream>>>(src, dst, cursor, csrc, E);

  // ---- layer 1: h1 = x @ W1 ; a1 = relu(gather + self + b1)
  k_gemm_wmma<<<Nn / 16, 256, 0, stream>>>(x, W1, h_buf, 384, 128);
  k_gather128<<<Nn / 8, 256, 0, stream>>>(rp, csrc, dis, h_buf, b1, agg_buf);

  // ---- layer 2: h2 = a1 @ W2 ; a2 = relu(gather + self + b2)
  k_gemm_wmma<<<Nn / 16, 128, 0, stream>>>(agg_buf, W2, h_buf, 128, 64);
  k_gather64<<<Nn / 8, 256, 0, stream>>>(rp, csrc, dis, h_buf, b2, agg_buf);

  // ---- fused mean-pool + linear head
  k_pool_lin<<<128, 256, 0, stream>>>(agg_buf, bat, Wlin, blin, out, Nn);
}